// NLRidge_26070451486882
// MI455X (gfx1250) — compile-verified
//
#include <hip/hip_runtime.h>
#include <math.h>

// ---------------- problem constants (from reference) ----------------
#define HH     256        // image H
#define WW     256        // image W
#define CC     3          // channels
#define PP     7          // patch size (P1 == P2 == 7)
#define NDIM   147        // C*P*P
#define HP     250        // H - P + 1
#define LPAT   62500      // HP*HP patch positions
#define RWIN   18         // search radius (w=37)
#define STEP   4
#define NG     4096       // 64*64 groups
#define M1     18
#define M2     55
#define LDG    64         // padded leading dim for Gram/theta
#define LDN    160        // padded feature dim (147 -> 160), 640B row stride
#define KMAXN  148        // ceil(147/4)*4

typedef __attribute__((ext_vector_type(2))) float v2f;
typedef __attribute__((ext_vector_type(8))) float v8f;

// ---------------- small utility kernels ----------------
__global__ void zero_kernel(float* __restrict__ p, int n) {
  int i = blockIdx.x * blockDim.x + threadIdx.x;
  if (i < n) p[i] = 0.f;
}

__global__ void mean3_kernel(const float* __restrict__ img, float* __restrict__ yb) {
  int i = blockIdx.x * blockDim.x + threadIdx.x;
  if (i < HH * WW)
    yb[i] = (img[i] + img[HH * WW + i] + img[2 * HH * WW + i]) * (1.0f / 3.0f);
}

// ---------------- block matching ----------------
// Faithful to the reference's inf-padded "align_corners" extended grid:
// extended coords 0..252; 250/251 are inf; 252 maps to row/col 249 but is
// only populated at positions that are multiples of STEP on the other axis.
__device__ __forceinline__ bool bm_valid(int er, int ec) {
  bool rin = (er >= 0 && er <= HP - 1);
  bool cin = (ec >= 0 && ec <= HP - 1);
  if (rin && cin) return true;
  if (er == 252 && cin) return (ec & 3) == 0;
  if (ec == 252 && rin) return (er & 3) == 0;
  if (er == 252 && ec == 252) return true;
  return false;
}

__global__ __launch_bounds__(64)
void block_matching_kernel(const float* __restrict__ yb, int* __restrict__ idxout, int M) {
  int c = blockIdx.x * blockDim.x + threadIdx.x;
  if (c >= NG) return;
  int ci = c >> 6, cj = c & 63;
  int ce_r = STEP * ci, ce_c = STEP * cj;           // extended center coords
  int crm = ce_r > HP - 1 ? HP - 1 : ce_r;          // mapped center patch row
  int ccm = ce_c > HP - 1 ? HP - 1 : ce_c;

  // keep the center patch rows warm (global_prefetch_b8 on gfx1250)
  #pragma unroll
  for (int u = 0; u < PP; ++u)
    __builtin_prefetch(yb + (crm + u) * WW + ccm, 0, 3);

  float dlist[M2];
  int   ilist[M2];
  int cnt = 0;

  for (int di = -RWIN; di <= RWIN; ++di) {
    int er = ce_r + di;
    for (int dj = -RWIN; dj <= RWIN; ++dj) {
      int ec = ce_c + dj;
      if (!bm_valid(er, ec)) continue;
      int pr = er > HP - 1 ? HP - 1 : er;
      int pc = ec > HP - 1 ? HP - 1 : ec;
      float d;
      if (di == 0 && dj == 0) {
        d = -3.0e38f;                                // center forced first
      } else {
        d = 0.f;
        for (int u = 0; u < PP; ++u) {
          const float* ra = yb + (crm + u) * WW + ccm;
          const float* rb = yb + (pr + u) * WW + pc;
          #pragma unroll
          for (int v = 0; v < PP; ++v) {
            float t = ra[v] - rb[v];
            d += t * t;
          }
        }
      }
      if (cnt == M && d >= dlist[M - 1]) continue;
      int pos = (cnt == M) ? (M - 1) : cnt++;
      while (pos > 0 && dlist[pos - 1] > d) {
        dlist[pos] = dlist[pos - 1];
        ilist[pos] = ilist[pos - 1];
        --pos;
      }
      int ir = crm + di; ir = ir < 0 ? 0 : (ir > HP - 1 ? HP - 1 : ir);
      int ic = ccm + dj; ic = ic < 0 ? 0 : (ic > HP - 1 ? HP - 1 : ic);
      dlist[pos] = d;
      ilist[pos] = ir * HP + ic;
    }
  }
  for (int t = 0; t < M; ++t) idxout[c * M + t] = ilist[t];
}

// ---------------- patch grouping into zero-padded [g][64][LDN] (rowsPad rows written) ----------------
__global__ void group_kernel(const float* __restrict__ img, const int* __restrict__ idx,
                             float* __restrict__ Y, int m, int rowsPad, long total) {
  long t = (long)blockIdx.x * blockDim.x + threadIdx.x;
  if (t >= total) return;
  int col = (int)(t % LDN);
  long rg = t / LDN;
  int r = (int)(rg % rowsPad);
  long g = rg / rowsPad;
  float v = 0.f;
  if (r < m && col < NDIM) {
    int fl = idx[g * m + r];
    int pr = fl / HP, pc = fl % HP;
    int ch = col / 49, rem = col % 49;
    int u = rem / PP, w = rem % PP;
    v = img[ch * HH * WW + (pr + u) * WW + (pc + w)];
  }
  Y[(size_t)g * 64 * LDN + (size_t)r * LDN + col] = v;
}

// ---------------- FP32 WMMA: Gram = Y * Y^T  (LDS-staged, per group) ----------------
// blockDim = 32 * tilesM * tilesM ; one wave per 16x16 output tile.
// Output slot is padded 64x64 -> fully unguarded stores (solve reads m x m only).
__global__ __launch_bounds__(512)
void gram_wmma_kernel(const float* __restrict__ Y, float* __restrict__ G,
                      int m, int tilesM) {
  __shared__ __align__(16) float sY[64 * LDN];      // 40 KB
  int g = blockIdx.x;
  int rowsPad = tilesM * 16;
  const float* Yg = Y + (size_t)g * 64 * LDN;
  int nquads = rowsPad * (LDN / 4);
  for (int i = threadIdx.x; i < nquads; i += blockDim.x)
    ((float4*)sY)[i] = ((const float4*)Yg)[i];
  __syncthreads();

  int wave = threadIdx.x >> 5;
  int lane = threadIdx.x & 31;
  int ti = wave / tilesM, tj = wave % tilesM;
  int half = lane >> 4, l16 = lane & 15;
  int arow = ti * 16 + l16;
  int bcol = tj * 16 + l16;
  v8f acc = {};
  for (int k = 0; k < KMAXN; k += 4) {
    int k0 = k + 2 * half;                          // ISA 16x4 f32 A/B layout
    float2 af = *(const float2*)&sY[arow * LDN + k0];
    float2 bf = *(const float2*)&sY[bcol * LDN + k0];
    v2f a, b;
    a[0] = af.x; a[1] = af.y;
    b[0] = bf.x; b[1] = bf.y;
    acc = __builtin_amdgcn_wmma_f32_16x16x4_f32(false, a, false, b, (short)0, acc,
                                                false, false);
  }
  float* Gg = G + (size_t)g * LDG * LDG;
  int Nc = tj * 16 + l16;
  #pragma unroll
  for (int r = 0; r < 8; ++r) {
    int Mr = ti * 16 + r + half * 8;
    Gg[Mr * LDG + Nc] = acc[r];                     // unguarded: padded slot
  }
}

// ---------------- per-group solve in LDS (Gauss-Jordan) ----------------
// stage 1: theta = solve(G, G - c*I)^T ; stage 2: theta = solve(G + c*I, G)^T
__global__ __launch_bounds__(64)
void solve_kernel(const float* __restrict__ Gbuf, const float* __restrict__ sig,
                  float* __restrict__ Theta, float* __restrict__ Wts, int m, int stage) {
  __shared__ float sA[M2 * M2];
  __shared__ float sB[M2 * M2];
  __shared__ float fcol[M2];
  int g = blockIdx.x, tid = threadIdx.x;
  const float* Gg = Gbuf + (size_t)g * LDG * LDG;
  float* Tg = Theta + (size_t)g * LDG * LDG;
  float s = sig[0];
  float creg = (float)NDIM * s * s;

  // zero-fill padded theta slot (so the GEMM K-loop needs no guards)
  for (int e = tid; e < LDG * LDG; e += blockDim.x) Tg[e] = 0.f;

  for (int e = tid; e < m * m; e += blockDim.x) {
    int i = e / m, j = e % m;
    float v = Gg[i * LDG + j];
    if (stage == 1) { sA[e] = v;                       sB[e] = (i == j) ? v - creg : v; }
    else            { sA[e] = (i == j) ? v + creg : v; sB[e] = v; }
  }
  __syncthreads();

  for (int k = 0; k < m; ++k) {
    float inv = 1.0f / sA[k * m + k];
    if (tid < m) fcol[tid] = sA[tid * m + k];
    __syncthreads();
    if (tid < m) { sA[k * m + tid] *= inv; sB[k * m + tid] *= inv; }
    __syncthreads();
    if (tid < m) {
      float ak = sA[k * m + tid];
      float bk = sB[k * m + tid];
      for (int i = 0; i < m; ++i) {
        if (i == k) continue;
        float f = fcol[i];
        sA[i * m + tid] -= f * ak;
        sB[i * m + tid] -= f * bk;
      }
    }
    __syncthreads();
  }
  // theta = X^T where X = sB ; store row-major, leading dim LDG
  for (int e = tid; e < m * m; e += blockDim.x) {
    int t = e / m, j = e % m;
    Tg[t * LDG + j] = sB[j * m + t];
  }
  if (tid < m) {
    float sum = 0.f;
    for (int j = 0; j < m; ++j) { float v = sB[j * m + tid]; sum += v * v; }
    Wts[g * m + tid] = 1.0f / sum;
  }
}

// ---------------- FP32 WMMA: Xhat = theta * Y  (LDS-staged) ----------------
// grid (NG, tilesM) ; blockDim = 320 (10 waves, one per 16-wide column tile).
// Output padded [64][LDN] -> fully unguarded stores.
__global__ __launch_bounds__(320)
void gemm_wmma_kernel(const float* __restrict__ Theta, const float* __restrict__ Y,
                      float* __restrict__ Xhat, int m, int tilesM) {
  __shared__ __align__(16) float sY[64 * LDN];      // 40 KB
  __shared__ __align__(16) float sT[16 * LDG];      // 4 KB (theta row stripe)
  int g = blockIdx.x, ti = blockIdx.y;
  int rowsPad = tilesM * 16;
  const float* Yg = Y + (size_t)g * 64 * LDN;
  for (int i = threadIdx.x; i < rowsPad * (LDN / 4); i += blockDim.x)
    ((float4*)sY)[i] = ((const float4*)Yg)[i];
  const float* Tg = Theta + (size_t)g * LDG * LDG + (size_t)ti * 16 * LDG;
  for (int i = threadIdx.x; i < 16 * (LDG / 4); i += blockDim.x)
    ((float4*)sT)[i] = ((const float4*)Tg)[i];
  __syncthreads();

  int tj = threadIdx.x >> 5;                        // 0..9 column tile
  int lane = threadIdx.x & 31;
  int half = lane >> 4, l16 = lane & 15;
  int bcol = tj * 16 + l16;                         // < 160
  v8f acc = {};
  for (int k = 0; k < rowsPad; k += 4) {
    int k0 = k + 2 * half;
    int k1 = k0 + 1;
    float2 af = *(const float2*)&sT[l16 * LDG + k0];
    v2f a, b;
    a[0] = af.x; a[1] = af.y;
    b[0] = sY[k0 * LDN + bcol];
    b[1] = sY[k1 * LDN + bcol];
    acc = __builtin_amdgcn_wmma_f32_16x16x4_f32(false, a, false, b, (short)0, acc,
                                                false, false);
  }
  float* Xg = Xhat + (size_t)g * 64 * LDN;
  #pragma unroll
  for (int r = 0; r < 8; ++r) {
    int Mr = ti * 16 + r + half * 8;
    Xg[(size_t)Mr * LDN + bcol] = acc[r];           // unguarded: padded slot
  }
}

// ---------------- aggregation: weighted scatter-add ----------------
// Xhat is padded [g][64][LDN]; only rows < m, cols < NDIM are read.
__global__ void scatter_kernel(const float* __restrict__ Xhat, const float* __restrict__ wts,
                               const int* __restrict__ idx, float* __restrict__ Xsum,
                               float* __restrict__ D1, int m) {
  int t = blockIdx.x * blockDim.x + threadIdx.x;
  if (t >= NG * m) return;
  float w = wts[t];
  int fl = idx[t];
  int g = t / m, r = t % m;
  const float* xh = Xhat + (size_t)g * 64 * LDN + (size_t)r * LDN;
  for (int j = 0; j < NDIM; ++j)
    atomicAdd(&Xsum[(size_t)j * LPAT + fl], xh[j] * w);
  atomicAdd(&D1[fl], w);
}

// ---------------- fold (overlap-add gather) + divide ----------------
__global__ void fold_div_kernel(const float* __restrict__ Xsum, const float* __restrict__ D1,
                                float* __restrict__ out) {
  int t = blockIdx.x * blockDim.x + threadIdx.x;
  if (t >= CC * HH * WW) return;
  int c = t / (HH * WW);
  int p = t % (HH * WW);
  int y = p / WW, x = p % WW;
  float num = 0.f, den = 0.f;
  for (int u = 0; u < PP; ++u) {
    int ry = y - u; if (ry < 0 || ry > HP - 1) continue;
    for (int v = 0; v < PP; ++v) {
      int rx = x - v; if (rx < 0 || rx > HP - 1) continue;
      int l = ry * HP + rx;
      num += Xsum[(size_t)(c * 49 + u * PP + v) * LPAT + l];
      den += D1[l];
    }
  }
  out[t] = num / den;
}

// ---------------- host orchestration ----------------
extern "C" void kernel_launch(void* const* d_in, const int* in_sizes, int n_in,
                              void* d_out, int out_size, void* d_ws, size_t ws_size,
                              hipStream_t stream) {
  const float* img = (const float*)d_in[0];
  const float* sig = (const float*)d_in[1];
  float* out = (float*)d_out;

  // bump-allocate workspace arena (floats)
  float* base  = (float*)d_ws;
  float* yb    = base;                       base += HH * WW;                   // 65536
  float* den1  = base;                       base += CC * HH * WW;              // 196608
  int*   idx   = (int*)base;                 base += NG * M2;                   // 225280
  float* Ybuf  = base;                       base += (size_t)NG * 64 * LDN;     // 41.9M
  float* Xbuf  = base;                       base += (size_t)NG * 64 * LDN;     // 41.9M
  float* Hbuf  = base;                       base += (size_t)NG * 64 * LDN;     // 41.9M (padded Xhat)
  float* Gram  = base;                       base += (size_t)NG * LDG * LDG;    // 16.8M
  float* Theta = base;                       base += (size_t)NG * LDG * LDG;    // 16.8M
  float* Wts   = base;                       base += NG * M2;
  float* Xsum  = base;                       base += (size_t)NDIM * LPAT;       // 9.19M
  float* D1    = base;                       base += LPAT;

  const int TB = 256;
  auto blocks = [](long n, int tb) { return (int)((n + tb - 1) / tb); };

  // ================= Stage 1 (m = M1 = 18, tilesM = 2) =================
  mean3_kernel<<<blocks(HH * WW, TB), TB, 0, stream>>>(img, yb);
  block_matching_kernel<<<blocks(NG, 64), 64, 0, stream>>>(yb, idx, M1);
  {
    long tot = (long)NG * 32 * LDN;
    group_kernel<<<blocks(tot, TB), TB, 0, stream>>>(img, idx, Ybuf, M1, 32, tot);
  }
  gram_wmma_kernel<<<NG, 32 * 2 * 2, 0, stream>>>(Ybuf, Gram, M1, 2);
  solve_kernel<<<NG, 64, 0, stream>>>(Gram, sig, Theta, Wts, M1, 1);
  gemm_wmma_kernel<<<dim3(NG, 2), 320, 0, stream>>>(Theta, Ybuf, Hbuf, M1, 2);
  zero_kernel<<<blocks((long)NDIM * LPAT, TB), TB, 0, stream>>>(Xsum, NDIM * LPAT);
  zero_kernel<<<blocks(LPAT, TB), TB, 0, stream>>>(D1, LPAT);
  scatter_kernel<<<blocks((long)NG * M1, TB), TB, 0, stream>>>(Hbuf, Wts, idx, Xsum, D1, M1);
  fold_div_kernel<<<blocks((long)CC * HH * WW, TB), TB, 0, stream>>>(Xsum, D1, den1);

  // ================= Stage 2 (m = M2 = 55, tilesM = 4) =================
  mean3_kernel<<<blocks(HH * WW, TB), TB, 0, stream>>>(den1, yb);
  block_matching_kernel<<<blocks(NG, 64), 64, 0, stream>>>(yb, idx, M2);
  {
    long tot = (long)NG * 64 * LDN;
    group_kernel<<<blocks(tot, TB), TB, 0, stream>>>(img,  idx, Ybuf, M2, 64, tot);
    group_kernel<<<blocks(tot, TB), TB, 0, stream>>>(den1, idx, Xbuf, M2, 64, tot);
  }
  gram_wmma_kernel<<<NG, 32 * 4 * 4, 0, stream>>>(Xbuf, Gram, M2, 4);
  solve_kernel<<<NG, 64, 0, stream>>>(Gram, sig, Theta, Wts, M2, 2);
  gemm_wmma_kernel<<<dim3(NG, 4), 320, 0, stream>>>(Theta, Ybuf, Hbuf, M2, 4);
  zero_kernel<<<blocks((long)NDIM * LPAT, TB), TB, 0, stream>>>(Xsum, NDIM * LPAT);
  zero_kernel<<<blocks(LPAT, TB), TB, 0, stream>>>(D1, LPAT);
  scatter_kernel<<<blocks((long)NG * M2, TB), TB, 0, stream>>>(Hbuf, Wts, idx, Xsum, D1, M2);
  fold_div_kernel<<<blocks((long)CC * HH * WW, TB), TB, 0, stream>>>(Xsum, D1, out);
}